// PCFG_base_41772851921136
// MI455X (gfx1250) — compile-verified
//
#include <hip/hip_runtime.h>
#include <stdint.h>

#define NN  256           // chart dimension (fixed by reference)
#define TPB 512           // 16 wave32 waves per workgroup

// Diagonal-major offset of diagonal d (1..NN-1) in the packed upper triangle:
// element (i, i+d) lives at chart[off_d(d) + i], i in [0, NN-d).
__device__ __forceinline__ int off_d(int d) {
  return (d - 1) * NN - ((d * (d - 1)) >> 1);
}

__global__ __launch_bounds__(TPB) void cky_inside_kernel(
    const float* __restrict__ marg,   // [B, NN, NN] f32
    const int*   __restrict__ lens,   // [B] i32
    float*       __restrict__ out) {  // [B] f32
  // Packed strict upper triangle: 32640 floats = 127.5 KB (fits CDNA5's 320KB LDS)
  __shared__ float chart[(NN * (NN - 1)) / 2];

  const int tid = threadIdx.x;
  const int b   = blockIdx.x;
  const float* mb = marg + (size_t)b * NN * NN;

  // ---------------------------------------------------------------------
  // Stage 1: async gather of marginals' upper triangle into the LDS chart.
  // chart[off(d)+i] = marg[b, i, i+d]. Uses CDNA5 GLOBAL_LOAD_ASYNC_TO_LDS
  // (ASYNCcnt) so the DMA overlaps per-lane address generation; no VGPR
  // round-trip. Diagonal 1 is exactly the reference's chart init.
  {
    int off = 0;
    for (int d = 1; d < NN; ++d) {
      const int len = NN - d;
      for (int i = tid; i < len; i += TPB) {
        uint32_t laddr = (uint32_t)(uintptr_t)(&chart[off + i]);        // LDS byte offset
        uint64_t gaddr = (uint64_t)(uintptr_t)(mb + (size_t)i * NN + i + d);
        asm volatile("global_load_async_to_lds_b32 %0, %1, off"
                     :: "v"(laddr), "v"(gaddr) : "memory");
      }
      off += len;
    }
    asm volatile("s_wait_asynccnt 0x0" ::: "memory");
  }
  __syncthreads();

  // ---------------------------------------------------------------------
  // Stage 2: CKY recursion over widths, chart resident in LDS.
  // Lanes are grouped G-per-span (G = pow2, <= 32, grows as the diagonal
  // shrinks) so the k-reduction stays parallel even for wide spans.
  for (int w = 2; w < NN; ++w) {
    const int n = NN - w;
    int G = 1;
    while (G < 32 && G * n < TPB) G <<= 1;   // scalar, uniform across block
    const int groups = TPB / G;
    const int gid    = tid / G;              // which span this group owns
    const int kl     = tid % G;              // lane's slice of the k-range
    const int offw   = off_d(w);

    for (int i = gid; i < n; i += groups) {
      // Online (streaming) logsumexp over this lane's k-slice.
      float m = -3.0e38f;   // finite sentinel: avoids (-inf - -inf) = NaN in merges
      float a = 0.0f;
      for (int k = kl; k <= w - 2; k += G) {
        const float y  = chart[off_d(k + 1)     + i];          // s[i, i+k+1]
        const float z  = chart[off_d(w - 1 - k) + i + k + 1];  // s[i+k+1, i+w]
        const float v  = y + z;
        const float mn = fmaxf(m, v);
        a = a * __expf(m - mn) + __expf(v - mn);
        m = mn;
      }
      // Butterfly merge of (m, a) across the G contiguous lanes of the group
      // (G divides 32, so groups never straddle a wave boundary).
      for (int s = G >> 1; s > 0; s >>= 1) {
        const float mo = __shfl_xor(m, s, 32);
        const float ao = __shfl_xor(a, s, 32);
        const float mn = fmaxf(m, mo);
        a = a * __expf(m - mn) + ao * __expf(mo - mn);
        m = mn;
      }
      if (kl == 0) {
        // chart cell currently holds marginals[b,i,i+w]; add the span score.
        chart[offw + i] += m + __logf(a);
      }
    }
    __syncthreads();   // diagonal w visible before width w+1
  }

  // ---------------------------------------------------------------------
  // Stage 3: emit s[b, 0, lens[b]]  (lens in [NN/2, NN), so diagonal >= 1)
  if (tid == 0) {
    const int L = lens[b];
    out[b] = chart[off_d(L)];   // i = 0 on diagonal L
  }
}

extern "C" void kernel_launch(void* const* d_in, const int* in_sizes, int n_in,
                              void* d_out, int out_size, void* d_ws, size_t ws_size,
                              hipStream_t stream) {
  const float* marg = (const float*)d_in[0];   // [B, 256, 256] f32
  const int*   lens = (const int*)d_in[1];     // [B] i32
  float*       out  = (float*)d_out;           // [B] f32
  const int B = in_sizes[1];                   // 64
  cky_inside_kernel<<<B, TPB, 0, stream>>>(marg, lens, out);
}